// DSP_33131377721365
// MI455X (gfx1250) — compile-verified
//
#include <hip/hip_runtime.h>
#include <hip/hip_bf16.h>

typedef __attribute__((ext_vector_type(2))) float v2f;
typedef __attribute__((ext_vector_type(4))) float v4f;
typedef __attribute__((ext_vector_type(8))) float v8f;

constexpr int BN = 4;
constexpr int CH = 32;
constexpr int H  = 720;
constexpr int W  = 720;
constexpr int W4 = W / 4;        // 180 float4 per row
constexpr int GP = W / 16;       // 45 16-pixel groups per row

// ---------------------------------------------------------------------------
// Pass 1: channel sum  s[b,h,w] = sum_c x[b,c,h,w]
// Streaming read-once of x (265 MB) with non-temporal hints so the small
// s/d intermediates stay resident in the 192 MB L2.
// ---------------------------------------------------------------------------
__global__ __launch_bounds__(256) void k_csum(const float* __restrict__ x,
                                              float* __restrict__ s) {
    int idx = blockIdx.x * blockDim.x + threadIdx.x;   // over BN*H*W4
    if (idx >= BN * H * W4) return;
    int w4 = idx % W4;
    int bh = idx / W4;                 // b*H + h
    int h  = bh % H;
    int b  = bh / H;
    const float* base = x + (((size_t)b * CH) * H + h) * (size_t)W + (size_t)w4 * 4;
    v4f acc = {0.f, 0.f, 0.f, 0.f};
#pragma unroll
    for (int c = 0; c < CH; ++c) {
        acc += __builtin_nontemporal_load((const v4f*)(base + (size_t)c * H * W));
    }
    *(v4f*)(s + (size_t)bh * W + (size_t)w4 * 4) = acc;
}

// ---------------------------------------------------------------------------
// Pass 2: diagonal 9-tap average (zero-padded by 4 on H and W)
// d[b,h,w] = (1/9) * sum_{t=0..8} s[b, h-4+t, w-4+t]
// Only ~17 MB of traffic, nearly all L2 hits.
// ---------------------------------------------------------------------------
__global__ __launch_bounds__(256) void k_diag(const float* __restrict__ s,
                                              float* __restrict__ d) {
    int idx = blockIdx.x * blockDim.x + threadIdx.x;   // over BN*H*W
    if (idx >= BN * H * W) return;
    int w = idx % W;
    int r = idx / W;
    int h = r % H;
    int b = r / H;
    float acc = 0.f;
#pragma unroll
    for (int t = 0; t < 9; ++t) {
        int hh = h + t - 4;
        int ww = w + t - 4;
        // clamp address into range, select zero for out-of-bounds taps
        int hc = hh < 0 ? 0 : (hh > H - 1 ? H - 1 : hh);
        int wc = ww < 0 ? 0 : (ww > W - 1 ? W - 1 : ww);
        float v = s[((size_t)b * H + hc) * W + wc];
        bool inb = (hh >= 0) & (hh < H) & (ww >= 0) & (ww < W);
        acc += inb ? v : 0.0f;
    }
    d[idx] = acc * (1.0f / 9.0f);
}

// ---------------------------------------------------------------------------
// Tiny pass: wsum[o,kh] = sum_i conv_w[o,i,kh,0]   (32x3 matrix)
// ---------------------------------------------------------------------------
__global__ void k_wsum(const float* __restrict__ cw, float* __restrict__ wsum) {
    int t = threadIdx.x;
    if (t >= CH * 3) return;
    int o  = t / 3;
    int kh = t % 3;
    float acc = 0.f;
    for (int i = 0; i < CH; ++i) acc += cw[((o * CH) + i) * 3 + kh];
    wsum[t] = acc;
}

// ---------------------------------------------------------------------------
// Pass 3: out[b,o,h,w] = sum_kh wsum[o,kh] * d[b, h-1+kh, w]
// Per 16-pixel group this is D(32x16) = Wsum(32x3) x Dtaps(3x16):
// two V_WMMA_F32_16X16X4_F32 per group (K=4, 4th tap zero).
//
// VGPR layouts per CDNA5 ISA 7.12.2 (wave32):
//   A 16x4 f32 (2 VGPRs): lanes 0-15 -> M=lane, VGPR j = K=j
//                         lanes 16-31 -> M=lane-16, VGPR j = K=j+2
//   B 4x16  f32 (2 VGPRs): lanes 0-15 -> K=j,  N=lane
//                          lanes 16-31 -> K=j+2, N=lane-16
//   C/D 16x16 f32 (8 VGPRs): VGPR v, lanes 0-15 -> M=v, N=lane
//                            lanes 16-31 -> M=v+8, N=lane-16
//
// All per-lane conditionals below are select-style (addresses valid for
// every lane) so the compiler emits v_cndmask instead of EXEC branches and
// the wave stays fully convergent up to the WMMAs.
// ---------------------------------------------------------------------------
__global__ __launch_bounds__(256) void k_conv_wmma(const float* __restrict__ d,
                                                   const float* __restrict__ wsum,
                                                   float* __restrict__ out) {
    int wave = (int)((blockIdx.x * blockDim.x + threadIdx.x) >> 5);
    int lane = threadIdx.x & 31;
    int total = BN * H * GP;
    if (wave >= total) return;          // wave-uniform: EXEC all-ones inside

    int wg = wave % GP;
    int bh = wave / GP;
    int h  = bh % H;
    int b  = bh / H;
    int w0 = wg * 16;

    int  lm = lane & 15;
    bool hiHalf = lane >= 16;

    // A matrices: wsum rows = output channels, K = kh tap (K=3 is zero pad).
    // Load unconditionally (addresses valid for all lanes), select after.
    int kA = hiHalf ? 2 : 0;            // K held by VGPR 0 in this lane half
    float a0k0 = wsum[lm * 3 + kA];
    float a0k1 = wsum[lm * 3 + 1];
    float a1k0 = wsum[(16 + lm) * 3 + kA];
    float a1k1 = wsum[(16 + lm) * 3 + 1];
    v2f A0, A1;
    A0[0] = a0k0;
    A0[1] = hiHalf ? 0.0f : a0k1;       // K=3 column is zero
    A1[0] = a1k0;
    A1[1] = hiHalf ? 0.0f : a1k1;

    // B matrix: rows are d at h-1+kh, cols are pixels w0..w0+15 (H zero pad).
    // Clamp the row index, load unconditionally, zero-select the pad rows.
    int hb0  = h + (hiHalf ? 1 : -1);   // tap kh = hiHalf ? 2 : 0
    int hb0c = hb0 < 0 ? 0 : (hb0 > H - 1 ? H - 1 : hb0);
    float bv0raw = d[((size_t)b * H + hb0c) * W + w0 + lm];
    float bv1raw = d[((size_t)b * H + h) * W + w0 + lm];     // tap kh=1
    bool  in0 = (hb0 >= 0) & (hb0 < H);
    v2f Bv;
    Bv[0] = in0 ? bv0raw : 0.0f;
    Bv[1] = hiHalf ? 0.0f : bv1raw;     // K=3 row is zero

    v8f c0 = {};
    v8f c1 = {};
    c0 = __builtin_amdgcn_wmma_f32_16x16x4_f32(false, A0, false, Bv,
                                               (short)0, c0, false, false);
    c1 = __builtin_amdgcn_wmma_f32_16x16x4_f32(false, A1, false, Bv,
                                               (short)0, c1, false, false);

    // Scatter D per the C/D layout; stores are two contiguous 64B runs each.
    size_t chanStride = (size_t)H * W;
    size_t outBase = ((size_t)b * CH) * chanStride + (size_t)h * W + w0 + lm;
    int oAdd = hiHalf ? 8 : 0;
#pragma unroll
    for (int v = 0; v < 8; ++v) {
        __builtin_nontemporal_store(c0[v], out + outBase + (size_t)(v + oAdd)      * chanStride);
        __builtin_nontemporal_store(c1[v], out + outBase + (size_t)(v + oAdd + 16) * chanStride);
    }
}

extern "C" void kernel_launch(void* const* d_in, const int* in_sizes, int n_in,
                              void* d_out, int out_size, void* d_ws, size_t ws_size,
                              hipStream_t stream) {
    const float* x  = (const float*)d_in[0];   // [4,32,720,720]
    const float* cw = (const float*)d_in[1];   // [32,32,3,1]
    float* out = (float*)d_out;                // [4,32,720,720]

    float* s    = (float*)d_ws;                         // BN*H*W floats
    float* dbuf = s + (size_t)BN * H * W;               // BN*H*W floats
    float* wsum = dbuf + (size_t)BN * H * W;            // 96 floats

    {   // Pass 1: channel sum
        int n = BN * H * W4;
        k_csum<<<(n + 255) / 256, 256, 0, stream>>>(x, s);
    }
    k_wsum<<<1, 96, 0, stream>>>(cw, wsum);
    {   // Pass 2: diagonal 9-tap average
        int n = BN * H * W;
        k_diag<<<(n + 255) / 256, 256, 0, stream>>>(s, dbuf);
    }
    {   // Pass 3: WMMA output-channel expansion + 3-tap H conv
        int waves  = BN * H * GP;      // 129600 waves, one 16-pixel group each
        int blocks = (waves + 7) / 8;  // 8 waves / 256-thread block
        k_conv_wmma<<<blocks, 256, 0, stream>>>(dbuf, wsum, out);
    }
}